// SimpleHGNLayer_9689446220172
// MI455X (gfx1250) — compile-verified
//
#include <hip/hip_runtime.h>

#define NHEAD 8
#define DOUT  64
#define HID   512   /* NHEAD*DOUT */
#define INF_  256   /* input feature dim */
#define NEG_SLOPE 0.2f

typedef float v2f __attribute__((ext_vector_type(2)));
typedef float v8f __attribute__((ext_vector_type(8)));

// ---------------------------------------------------------------------------
// Tiny precompute: contract W with a_l / a_r, W_e with a_e, and fold edge_emb
//   w_l[k][h] = sum_d W[k, h*64+d] * a_l[h,d]          (k < 256)
//   w_r[k][h] = sum_d W[k, h*64+d] * a_r[h,d]
//   w_e[k][h] = sum_f W_e[k, h*64+f] * a_e[h,f]        (k < 64)
//   h_e_type[t][h] = sum_k edge_emb[t,k] * w_e[k][h]   (t < ntypes)
// ---------------------------------------------------------------------------
__global__ __launch_bounds__(256)
void precompute_kernel(const float* __restrict__ W, const float* __restrict__ We,
                       const float* __restrict__ a_l, const float* __restrict__ a_r,
                       const float* __restrict__ a_e, const float* __restrict__ edge_emb,
                       float* __restrict__ w_l, float* __restrict__ w_r,
                       float* __restrict__ w_e, float* __restrict__ h_e_type,
                       int ntypes) {
  int t = threadIdx.x;  // 256 threads, one per k-row of W
  #pragma unroll
  for (int h = 0; h < NHEAD; ++h) {
    float sl = 0.f, sr = 0.f;
    for (int d = 0; d < DOUT; ++d) {
      float w = W[t * HID + h * DOUT + d];
      sl += w * a_l[h * DOUT + d];
      sr += w * a_r[h * DOUT + d];
    }
    w_l[t * NHEAD + h] = sl;
    w_r[t * NHEAD + h] = sr;
  }
  if (t < 64) {
    #pragma unroll
    for (int h = 0; h < NHEAD; ++h) {
      float s = 0.f;
      for (int f = 0; f < 64; ++f)
        s += We[t * HID + h * 64 + f] * a_e[h * 64 + f];
      w_e[t * NHEAD + h] = s;
    }
  }
  __syncthreads();
  if (t < ntypes * NHEAD) {
    int ty = t / NHEAD, h = t % NHEAD;
    float s = 0.f;
    for (int k = 0; k < 64; ++k) s += edge_emb[ty * 64 + k] * w_e[k * NHEAD + h];
    h_e_type[t] = s;
  }
}

// ---------------------------------------------------------------------------
// h = X @ wred   with X: [n,256], wred: [256,8] -> out [n,8]
// One wave per row; lane owns 8 consecutive k, weights preloaded to registers.
// ---------------------------------------------------------------------------
__global__ __launch_bounds__(256)
void rowdot_kernel(const float* __restrict__ X, const float* __restrict__ wred,
                   float* __restrict__ outv, int nrows) {
  int wave = threadIdx.x >> 5, lane = threadIdx.x & 31;
  float wr[8][NHEAD];
  #pragma unroll
  for (int j = 0; j < 8; ++j) {
    const float* wp = wred + (lane * 8 + j) * NHEAD;
    #pragma unroll
    for (int h = 0; h < NHEAD; ++h) wr[j][h] = wp[h];
  }
  for (int row = blockIdx.x * 8 + wave; row < nrows; row += gridDim.x * 8) {
    const float* xr = X + (size_t)row * INF_ + lane * 8;
    float acc[NHEAD];
    #pragma unroll
    for (int h = 0; h < NHEAD; ++h) acc[h] = 0.f;
    #pragma unroll
    for (int j = 0; j < 8; ++j) {
      float x = xr[j];
      #pragma unroll
      for (int h = 0; h < NHEAD; ++h) acc[h] += x * wr[j][h];
    }
    #pragma unroll
    for (int h = 0; h < NHEAD; ++h) {
      #pragma unroll
      for (int off = 16; off > 0; off >>= 1)
        acc[h] += __shfl_xor(acc[h], off, 32);
    }
    if (lane == 0) {
      #pragma unroll
      for (int h = 0; h < NHEAD; ++h) outv[(size_t)row * NHEAD + h] = acc[h];
    }
  }
}

// ---------------------------------------------------------------------------
// FP32 WMMA GEMM:  Y[n,512] = X[n,256] @ W[256,512]
// Block 256 thr (8 waves), tile 128x64, K-tile 32 in LDS, wave = 32x32 subtile
// (2x2 of v_wmma_f32_16x16x4_f32 accumulators).
// ---------------------------------------------------------------------------
__global__ __launch_bounds__(256)
void gemm_wmma_kernel(const float* __restrict__ X, const float* __restrict__ Wm,
                      float* __restrict__ Y, int nrows) {
  constexpr int TM = 128, TN = 64, TK = 32, LDA = 36, LDB = 80;
  __shared__ __align__(16) float lA[TM * LDA];   // 18.0 KB
  __shared__ __align__(16) float lB[TK * LDB];   // 10.0 KB
  const int tid  = threadIdx.x;
  const int wave = tid >> 5, lane = tid & 31;
  const int half = lane >> 4, lm = lane & 15;
  const int m0 = (wave >> 1) * 32, n0 = (wave & 1) * 32;
  const int rowBase = blockIdx.x * TM;
  const int colBase = blockIdx.y * TN;

  v8f acc00 = {0.f,0.f,0.f,0.f,0.f,0.f,0.f,0.f};
  v8f acc01 = acc00, acc10 = acc00, acc11 = acc00;

  for (int kb = 0; kb < INF_; kb += TK) {
    // Stage A tile (128x32) : 1024 float4, 4 per thread
    #pragma unroll
    for (int i = 0; i < 4; ++i) {
      int idx = tid + i * 256;
      int r = idx >> 3;
      int c = (idx & 7) << 2;
      int grow = rowBase + r;
      float4 v = make_float4(0.f, 0.f, 0.f, 0.f);
      if (grow < nrows) v = *(const float4*)(X + (size_t)grow * INF_ + kb + c);
      *(float4*)(&lA[r * LDA + c]) = v;
    }
    // Stage B tile (32x64) : 512 float4, 2 per thread
    #pragma unroll
    for (int i = 0; i < 2; ++i) {
      int idx = tid + i * 256;
      int kr = idx >> 4;
      int c = (idx & 15) << 2;
      float4 v = *(const float4*)(Wm + (size_t)(kb + kr) * HID + colBase + c);
      *(float4*)(&lB[kr * LDB + c]) = v;
    }
    __syncthreads();

    #pragma unroll
    for (int kt = 0; kt < TK; kt += 4) {
      const int ak = kt + 2 * half;  // K halves per ISA A-frag layout
      v2f a0, a1, b0, b1;
      a0[0] = lA[(m0 + lm) * LDA + ak];
      a0[1] = lA[(m0 + lm) * LDA + ak + 1];
      a1[0] = lA[(m0 + 16 + lm) * LDA + ak];
      a1[1] = lA[(m0 + 16 + lm) * LDA + ak + 1];
      b0[0] = lB[ak * LDB + n0 + lm];
      b0[1] = lB[(ak + 1) * LDB + n0 + lm];
      b1[0] = lB[ak * LDB + n0 + 16 + lm];
      b1[1] = lB[(ak + 1) * LDB + n0 + 16 + lm];
      acc00 = __builtin_amdgcn_wmma_f32_16x16x4_f32(false, a0, false, b0, (short)0, acc00, false, false);
      acc01 = __builtin_amdgcn_wmma_f32_16x16x4_f32(false, a0, false, b1, (short)0, acc01, false, false);
      acc10 = __builtin_amdgcn_wmma_f32_16x16x4_f32(false, a1, false, b0, (short)0, acc10, false, false);
      acc11 = __builtin_amdgcn_wmma_f32_16x16x4_f32(false, a1, false, b1, (short)0, acc11, false, false);
    }
    __syncthreads();
  }

  // C/D layout: VGPR r, lane l -> M = r + 8*(l>>4), N = l&15
  #pragma unroll
  for (int r = 0; r < 8; ++r) {
    int row0 = rowBase + m0 + r + 8 * half;
    int row1 = row0 + 16;
    int col0 = colBase + n0 + lm;
    if (row0 < nrows) {
      Y[(size_t)row0 * HID + col0]      = acc00[r];
      Y[(size_t)row0 * HID + col0 + 16] = acc01[r];
    }
    if (row1 < nrows) {
      Y[(size_t)row1 * HID + col0]      = acc10[r];
      Y[(size_t)row1 * HID + col0 + 16] = acc11[r];
    }
  }
}

// ---------------------------------------------------------------------------
// CSR build: count -> scan -> fill -> per-node sort (deterministic edge order)
// ---------------------------------------------------------------------------
__global__ void count_kernel(const int* __restrict__ head_ind, int* __restrict__ cnt, int E_) {
  int e = blockIdx.x * 256 + threadIdx.x;
  if (e < E_) atomicAdd(&cnt[head_ind[e]], 1);
}

__global__ __launch_bounds__(1024)
void scan_kernel(const int* __restrict__ cnt, int* __restrict__ ptr,
                 int* __restrict__ run, int n) {
  __shared__ int sh[1024];
  int t = threadIdx.x;
  int carry = 0;
  for (int base = 0; base < n; base += 1024) {
    int v = (base + t < n) ? cnt[base + t] : 0;
    sh[t] = v;
    __syncthreads();
    for (int off = 1; off < 1024; off <<= 1) {
      int add = (t >= off) ? sh[t - off] : 0;
      __syncthreads();
      sh[t] += add;
      __syncthreads();
    }
    if (base + t < n) {
      int p = carry + sh[t] - v;  // exclusive
      ptr[base + t] = p;
      run[base + t] = p;
    }
    carry += sh[1023];
    __syncthreads();
  }
  if (t == 0) ptr[n] = carry;
}

__global__ void fill_kernel(const int* __restrict__ head_ind, int* __restrict__ run,
                            int* __restrict__ csr_edges, int E_) {
  int e = blockIdx.x * 256 + threadIdx.x;
  if (e < E_) {
    int pos = atomicAdd(&run[head_ind[e]], 1);
    csr_edges[pos] = e;
  }
}

__global__ void sort_kernel(const int* __restrict__ ptr, int* __restrict__ edges, int n) {
  int i = blockIdx.x * 256 + threadIdx.x;
  if (i >= n) return;
  int s = ptr[i], e = ptr[i + 1];
  for (int a = s + 1; a < e; ++a) {  // insertion sort, avg degree ~16
    int key = edges[a];
    int b = a - 1;
    while (b >= s && edges[b] > key) { edges[b + 1] = edges[b]; --b; }
    edges[b + 1] = key;
  }
}

// ---------------------------------------------------------------------------
// Fused per-destination softmax + aggregation. One wave per head node.
// Lane l owns output columns {l + 32j : j=0..15}; head of column = j>>1.
// ---------------------------------------------------------------------------
__global__ __launch_bounds__(256)
void agg_kernel(const int* __restrict__ ptr, const int* __restrict__ edges,
                const int* __restrict__ tail_ind, const int* __restrict__ etype,
                const float* __restrict__ h_l, const float* __restrict__ h_r,
                const float* __restrict__ he, const float* __restrict__ hT,
                float* __restrict__ out, float* __restrict__ att, int NH_) {
  int wave = threadIdx.x >> 5;
  int lane = threadIdx.x & 31;
  int i = blockIdx.x * 8 + wave;
  if (i >= NH_) return;
  int s = ptr[i], e = ptr[i + 1];

  float hl[NHEAD];
  #pragma unroll
  for (int h = 0; h < NHEAD; ++h) hl[h] = h_l[(size_t)i * NHEAD + h];

  float m[NHEAD];
  #pragma unroll
  for (int h = 0; h < NHEAD; ++h) m[h] = -3.402823466e+38f;

  // pass 1: per-head max of leaky-relu logits
  for (int idx = s; idx < e; ++idx) {
    int ce = edges[idx];
    int t = tail_ind[ce], ty = etype[ce];
    #pragma unroll
    for (int h = 0; h < NHEAD; ++h) {
      float lg = hl[h] + h_r[(size_t)t * NHEAD + h] + he[ty * NHEAD + h];
      lg = lg > 0.f ? lg : NEG_SLOPE * lg;
      m[h] = fmaxf(m[h], lg);
    }
  }

  // pass 2: exp-sum + unnormalized att + weighted aggregation
  float sden[NHEAD];
  #pragma unroll
  for (int h = 0; h < NHEAD; ++h) sden[h] = 0.f;
  float acc[16];
  #pragma unroll
  for (int j = 0; j < 16; ++j) acc[j] = 0.f;

  for (int idx = s; idx < e; ++idx) {
    int ce = edges[idx];
    int t = tail_ind[ce], ty = etype[ce];
    float w[NHEAD];
    #pragma unroll
    for (int h = 0; h < NHEAD; ++h) {
      float lg = hl[h] + h_r[(size_t)t * NHEAD + h] + he[ty * NHEAD + h];
      lg = lg > 0.f ? lg : NEG_SLOPE * lg;
      w[h] = __expf(lg - m[h]);
      sden[h] += w[h];
    }
    if (lane == 0) {
      #pragma unroll
      for (int h = 0; h < NHEAD; ++h) att[(size_t)ce * NHEAD + h] = w[h];
    }
    const float* row = hT + (size_t)t * HID;
    #pragma unroll
    for (int j = 0; j < 16; ++j)
      acc[j] += w[j >> 1] * row[lane + 32 * j];
  }

  float inv[NHEAD];
  #pragma unroll
  for (int h = 0; h < NHEAD; ++h) inv[h] = sden[h] > 0.f ? 1.f / sden[h] : 0.f;

  // pass 3: normalize stored att
  if (lane == 0) {
    for (int idx = s; idx < e; ++idx) {
      int ce = edges[idx];
      #pragma unroll
      for (int h = 0; h < NHEAD; ++h) att[(size_t)ce * NHEAD + h] *= inv[h];
    }
  }

  #pragma unroll
  for (int j = 0; j < 16; ++j)
    out[(size_t)i * HID + lane + 32 * j] = acc[j] * inv[j >> 1];
}

// ---------------------------------------------------------------------------
extern "C" void kernel_launch(void* const* d_in, const int* in_sizes, int n_in,
                              void* d_out, int out_size, void* d_ws, size_t ws_size,
                              hipStream_t stream) {
  const float* head_feature = (const float*)d_in[0];
  const float* tail_feature = (const float*)d_in[1];
  const int*   head_ind     = (const int*)d_in[2];
  const int*   tail_ind     = (const int*)d_in[3];
  const int*   tmp_edge     = (const int*)d_in[4];
  const float* edge_emb     = (const float*)d_in[5];
  const float* W            = (const float*)d_in[6];
  const float* We           = (const float*)d_in[7];
  const float* a_l          = (const float*)d_in[8];
  const float* a_r          = (const float*)d_in[9];
  const float* a_e          = (const float*)d_in[10];

  const int NH = in_sizes[0] / INF_;
  const int NT = in_sizes[1] / INF_;
  const int E  = in_sizes[2];
  const int NTYPES = in_sizes[5] / 64;

  float* out = (float*)d_out;
  float* att = out + (size_t)NH * HID;

  char* wp = (char*)d_ws;
  size_t off = 0;
  auto take = [&](size_t bytes) -> char* {
    char* p = wp + off;
    off = (off + bytes + 255) & ~(size_t)255;
    return p;
  };
  float* hTail    = (float*)take((size_t)NT * HID * sizeof(float));   // 204.8 MB
  float* h_l      = (float*)take((size_t)NH * NHEAD * sizeof(float));
  float* h_r      = (float*)take((size_t)NT * NHEAD * sizeof(float));
  float* w_l      = (float*)take((size_t)INF_ * NHEAD * sizeof(float));
  float* w_r      = (float*)take((size_t)INF_ * NHEAD * sizeof(float));
  float* w_e      = (float*)take((size_t)64 * NHEAD * sizeof(float));
  float* h_e_type = (float*)take((size_t)NTYPES * NHEAD * sizeof(float));
  int* cnt        = (int*)take((size_t)NH * sizeof(int));
  int* csr_ptr    = (int*)take((size_t)(NH + 1) * sizeof(int));
  int* run        = (int*)take((size_t)NH * sizeof(int));
  int* csr_edges  = (int*)take((size_t)E * sizeof(int));

  hipMemsetAsync(cnt, 0, (size_t)NH * sizeof(int), stream);

  precompute_kernel<<<1, 256, 0, stream>>>(W, We, a_l, a_r, a_e, edge_emb,
                                           w_l, w_r, w_e, h_e_type, NTYPES);
  rowdot_kernel<<<1024, 256, 0, stream>>>(head_feature, w_l, h_l, NH);
  rowdot_kernel<<<1024, 256, 0, stream>>>(tail_feature, w_r, h_r, NT);

  dim3 gg((NT + 127) / 128, HID / 64);
  gemm_wmma_kernel<<<gg, 256, 0, stream>>>(tail_feature, W, hTail, NT);

  count_kernel<<<(E + 255) / 256, 256, 0, stream>>>(head_ind, cnt, E);
  scan_kernel<<<1, 1024, 0, stream>>>(cnt, csr_ptr, run, NH);
  fill_kernel<<<(E + 255) / 256, 256, 0, stream>>>(head_ind, run, csr_edges, E);
  sort_kernel<<<(NH + 255) / 256, 256, 0, stream>>>(csr_ptr, csr_edges, NH);

  agg_kernel<<<(NH + 7) / 8, 256, 0, stream>>>(csr_ptr, csr_edges, tail_ind, tmp_edge,
                                               h_l, h_r, h_e_type, hTail, out, att, NH);

  (void)n_in; (void)out_size; (void)ws_size;
}